// MANN_23965917511952
// MI455X (gfx1250) — compile-verified
//
#include <hip/hip_runtime.h>
#include <hip/hip_bf16.h>
#include <float.h>

typedef __attribute__((ext_vector_type(2))) float v2f;
typedef __attribute__((ext_vector_type(4))) float v4f;
typedef __attribute__((ext_vector_type(8))) float v8f;

// gfx1250 async global->LDS path (ASYNCcnt-tracked), with a safe fallback if
// this toolchain doesn't declare the builtins.
#if defined(__HIP_DEVICE_COMPILE__) &&                                   \
    __has_builtin(__builtin_amdgcn_global_load_async_to_lds_b128) &&     \
    __has_builtin(__builtin_amdgcn_s_wait_asynccnt)
#define USE_ASYNC_LDS 1
// Builtin signature (from hipcc diagnostic): param0 = int4 AS1(global)*,
// param1 = LDS-side pointer, then imm offset + imm cpol.
typedef int v4i_ld __attribute__((vector_size(16)));
typedef __attribute__((address_space(1))) v4i_ld* async_gptr_t;
typedef __attribute__((address_space(3))) v4i_ld* async_lptr_t;
#else
#define USE_ASYNC_LDS 0
#endif

// ---------------------------------------------------------------------------
// Kernel 1: row-wise max over L (streaming, ~440MB total => sets the ~19us
// HBM roofline at 23.3 TB/s), then LayerNorm over H=256.
// One block per row, 256 threads. Thread t = (column-group cg = t&63) x
// (L-slice ls = t>>6): b128 non-temporal loads, 512B contiguous per wave per
// step, 4x fewer VMEM instructions than b32-per-lane. NT keeps the one-shot
// stream from evicting the L2-resident working buffers.
// ---------------------------------------------------------------------------
__global__ __launch_bounds__(256) void rowmax_ln_kernel(
    const float* __restrict__ x, const float* __restrict__ gamma,
    const float* __restrict__ beta, float* __restrict__ out, int L) {
  const int t = threadIdx.x;
  const int row = blockIdx.x;
  const int cg = t & 63;  // column group: columns 4*cg .. 4*cg+3
  const int ls = t >> 6;  // L slice 0..3

  const v4f* p = (const v4f*)(x + (size_t)row * (size_t)L * 256) + cg;
  v4f m = {-FLT_MAX, -FLT_MAX, -FLT_MAX, -FLT_MAX};
  for (int l = ls; l < L; l += 4) {
    __builtin_prefetch(p + (size_t)(l + 16) * 64, 0, 0);  // global_prefetch_b8
    const v4f v = __builtin_nontemporal_load(p + (size_t)l * 64);
    m.x = fmaxf(m.x, v.x);
    m.y = fmaxf(m.y, v.y);
    m.z = fmaxf(m.z, v.z);
    m.w = fmaxf(m.w, v.w);
  }

  __shared__ v4f smax4[256];
  __shared__ float sm[256];
  __shared__ float red[256];
  smax4[t] = m;
  __syncthreads();
  if (t < 64) {  // combine the 4 L-slices per column group
    const v4f a = smax4[t], b = smax4[t + 64], c = smax4[t + 128],
              d = smax4[t + 192];
    sm[4 * t + 0] = fmaxf(fmaxf(a.x, b.x), fmaxf(c.x, d.x));
    sm[4 * t + 1] = fmaxf(fmaxf(a.y, b.y), fmaxf(c.y, d.y));
    sm[4 * t + 2] = fmaxf(fmaxf(a.z, b.z), fmaxf(c.z, d.z));
    sm[4 * t + 3] = fmaxf(fmaxf(a.w, b.w), fmaxf(c.w, d.w));
  }
  __syncthreads();

  const float mval = sm[t];  // this thread's column value
  // mean (deterministic tree reduce)
  red[t] = mval;
  __syncthreads();
  for (int s = 128; s > 0; s >>= 1) {
    if (t < s) red[t] += red[t + s];
    __syncthreads();
  }
  const float mu = red[0] * (1.0f / 256.0f);
  __syncthreads();
  // variance (two-pass, matches jnp.var = mean((x-mu)^2))
  const float d = mval - mu;
  red[t] = d * d;
  __syncthreads();
  for (int s = 128; s > 0; s >>= 1) {
    if (t < s) red[t] += red[t + s];
    __syncthreads();
  }
  const float var = red[0] * (1.0f / 256.0f);
  out[(size_t)row * 256 + t] = d * rsqrtf(var + 1e-5f) * gamma[t] + beta[t];
}

// ---------------------------------------------------------------------------
// Kernel 2: fused MANN tile: logits = Ktile(16x256) @ mem^T(256x128) via
// V_WMMA_F32_16X16X4_F32, row softmax, R = W(16x128) @ mem(128x256) via WMMA,
// out = R + K. 8 waves per block: wave w owns logit cols [16w,16w+16) in
// GEMM1 and out cols [32w,32w+32) (two accumulators) in GEMM2.
// WMMA f32 lane map: A: M=l%16, K=2*(l/16)+v ; B: N=l%16, K=2*(l/16)+v ;
// C/D: N=l%16, M=v+8*(l/16).
// K tile staged via GLOBAL_LOAD_ASYNC_TO_LDS_B128 when available.
// ---------------------------------------------------------------------------
__global__ __launch_bounds__(256) void mann_tile_kernel(
    const float* __restrict__ Kin, const float* __restrict__ mem,
    float* __restrict__ out, float* __restrict__ Rbuf, int memRows) {
  __shared__ float ldsK[16 * 256];  // 16KB: K tile
  __shared__ float ldsW[16 * 128];  // 8KB : logits -> softmax weights
  const int tid = threadIdx.x;
  const int row0 = blockIdx.x * 16;

  // stage the 16x256 K tile into LDS
#if USE_ASYNC_LDS
  {
    const float* src = Kin + (size_t)row0 * 256 + tid * 4;
    float* dst = ldsK + tid * 4;
    #pragma unroll
    for (int c = 0; c < 4; ++c) {  // 4 x 4KB chunks, 16B per lane each
      __builtin_amdgcn_global_load_async_to_lds_b128(
          (async_gptr_t)(src + c * 1024), (async_lptr_t)(dst + c * 1024),
          /*offset=*/0, /*cpol=*/0);
    }
    __builtin_amdgcn_s_wait_asynccnt(0);
  }
#else
  {
    const v4f* src = (const v4f*)(Kin + (size_t)row0 * 256);
    v4f* dst = (v4f*)ldsK;
    for (int i = tid; i < 16 * 64; i += 256) dst[i] = src[i];
  }
#endif
  __syncthreads();

  const int wave = tid >> 5;
  const int lane = tid & 31;
  const int lm = lane & 15;   // N index inside 16x16 tile
  const int lh = lane >> 4;   // half-wave select

  // ---- GEMM1: 16x16 logits tile for mem slots [n0, n0+16)
  const int n0 = wave * 16;
  v8f acc = {};
  const float* bbase = mem + (size_t)(n0 + lm) * 256;  // B = mem^T
  for (int kb = 0; kb < 256; kb += 4) {
    const int kk = kb + 2 * lh;
    v2f a;
    a.x = ldsK[lm * 256 + kk];
    a.y = ldsK[lm * 256 + kk + 1];
    v2f b;
    b.x = bbase[kk];
    b.y = bbase[kk + 1];
    acc = __builtin_amdgcn_wmma_f32_16x16x4_f32(false, a, false, b, (short)0,
                                                acc, false, false);
  }
  #pragma unroll
  for (int v = 0; v < 8; ++v) ldsW[(v + 8 * lh) * 128 + n0 + lm] = acc[v];
  __syncthreads();

  // ---- softmax over the 128 mem slots, one thread per row (16 rows)
  if (tid < 16) {
    float* w = ldsW + tid * 128;
    float mx = w[0];
    for (int j = 1; j < 128; ++j) mx = fmaxf(mx, w[j]);
    float s = 0.f;
    for (int j = 0; j < 128; ++j) {
      const float e = __expf(w[j] - mx);
      w[j] = e;
      s += e;
    }
    const float inv = 1.f / s;
    for (int j = 0; j < 128; ++j) w[j] *= inv;
  }
  __syncthreads();

  // ---- GEMM2: R(16x256) = W(16x128) @ mem(128x256); this wave: cols [h0,h0+32)
  const int h0 = wave * 32;
  v8f r0 = {};
  v8f r1 = {};
  for (int kb = 0; kb < 128; kb += 4) {
    const int kk = kb + 2 * lh;
    v2f a;
    a.x = ldsW[lm * 128 + kk];
    a.y = ldsW[lm * 128 + kk + 1];
    v2f b0, b1;
    b0.x = mem[(size_t)kk * 256 + h0 + lm];
    b0.y = mem[(size_t)(kk + 1) * 256 + h0 + lm];
    b1.x = mem[(size_t)kk * 256 + h0 + 16 + lm];
    b1.y = mem[(size_t)(kk + 1) * 256 + h0 + 16 + lm];
    r0 = __builtin_amdgcn_wmma_f32_16x16x4_f32(false, a, false, b0, (short)0,
                                               r0, false, false);
    r1 = __builtin_amdgcn_wmma_f32_16x16x4_f32(false, a, false, b1, (short)0,
                                               r1, false, false);
  }

  // ---- store out = R + K ; stash raw R for rows < memRows (mem write-back)
  const bool keepR = (Rbuf != nullptr) && (row0 < memRows);
  #pragma unroll
  for (int v = 0; v < 8; ++v) {
    const int m = v + 8 * lh;
    const int c0 = h0 + lm;
    const int c1 = h0 + 16 + lm;
    out[(size_t)(row0 + m) * 256 + c0] = r0[v] + ldsK[m * 256 + c0];
    out[(size_t)(row0 + m) * 256 + c1] = r1[v] + ldsK[m * 256 + c1];
    if (keepR) {
      Rbuf[(size_t)(row0 + m) * 256 + c0] = r0[v];
      Rbuf[(size_t)(row0 + m) * 256 + c1] = r1[v];
    }
  }
}

// ---------------------------------------------------------------------------
// Kernel 3: mem1[k] = R[rank(k)], rank = stable-argsort position of mem_weight
// (ties broken by index, matching jnp.argsort stable semantics). Integer LDS
// atomics only -> bitwise deterministic across graph replays.
// ---------------------------------------------------------------------------
__global__ __launch_bounds__(256) void mem_update_kernel(
    const float* __restrict__ mw, const float* __restrict__ Rbuf,
    float* __restrict__ mem1) {
  const int k = blockIdx.x;   // mem slot 0..127
  const int t = threadIdx.x;  // 0..255
  __shared__ float smw[128];
  __shared__ int scnt;
  if (t < 128) smw[t] = mw[t];
  if (t == 0) scnt = 0;
  __syncthreads();
  const float mk = smw[k];
  if (t < 128) {
    const float mj = smw[t];
    if (mj < mk || (mj == mk && t < k)) atomicAdd(&scnt, 1);
  }
  __syncthreads();
  const int rank = scnt;
  mem1[(size_t)k * 256 + t] = Rbuf[(size_t)rank * 256 + t];
}

// ---------------------------------------------------------------------------
// Kernel 4: prototypes = mean over K shots. Block per (episode, class) row.
// ---------------------------------------------------------------------------
__global__ __launch_bounds__(256) void proto_kernel(
    const float* __restrict__ S2, float* __restrict__ proto, int Kshot) {
  const int o = blockIdx.x;  // b*N + n
  const int t = threadIdx.x;
  float s = 0.f;
  for (int k = 0; k < Kshot; ++k) s += S2[(size_t)(o * Kshot + k) * 256 + t];
  proto[(size_t)o * 256 + t] = s / (float)Kshot;
}

// ---------------------------------------------------------------------------
// Kernel 5: logits[b,j,c] = -||q - proto||^2, minn concat, argmax.
// Deterministic shared-memory tree reduction per class.
// ---------------------------------------------------------------------------
__global__ __launch_bounds__(256) void logits_kernel(
    const float* __restrict__ Q2, const float* __restrict__ proto,
    float* __restrict__ out_logits, float* __restrict__ out_pred, int nqPerEp,
    int nClasses) {
  const int bj = blockIdx.x;
  const int t = threadIdx.x;
  const int b = bj / nqPerEp;
  __shared__ float red[256];
  __shared__ float lg[16];
  const float qv = Q2[(size_t)bj * 256 + t];
  for (int c = 0; c < nClasses; ++c) {
    const float d = qv - proto[(size_t)(b * nClasses + c) * 256 + t];
    red[t] = d * d;
    __syncthreads();
    for (int s = 128; s > 0; s >>= 1) {
      if (t < s) red[t] += red[t + s];
      __syncthreads();
    }
    if (t == 0) lg[c] = -red[0];
    __syncthreads();
  }
  if (t == 0) {
    float mn = lg[0];
    for (int c = 1; c < nClasses; ++c) mn = fminf(mn, lg[c]);
    lg[nClasses] = mn - 1.0f;
    int best = 0;
    float bv = lg[0];
    for (int c = 1; c <= nClasses; ++c)
      if (lg[c] > bv) { bv = lg[c]; best = c; }  // first-occurrence argmax
    for (int c = 0; c <= nClasses; ++c)
      out_logits[(size_t)bj * (nClasses + 1) + c] = lg[c];
    out_pred[bj] = (float)best;
  }
}

// ---------------------------------------------------------------------------
extern "C" void kernel_launch(void* const* d_in, const int* in_sizes, int n_in,
                              void* d_out, int out_size, void* d_ws,
                              size_t ws_size, hipStream_t stream) {
  (void)n_in; (void)out_size; (void)ws_size;
  const float* support = (const float*)d_in[0];
  const float* query   = (const float*)d_in[1];
  const float* gamma   = (const float*)d_in[2];
  const float* beta    = (const float*)d_in[3];
  const float* mem0    = (const float*)d_in[4];
  const float* mw      = (const float*)d_in[5];

  const int L = 128, H = 256, M = 128;   // fixed by setup_inputs
  const int N = 10, Kshot = 5;
  const int rows_s = in_sizes[0] / (L * H);     // 1600
  const int rows_q = in_sizes[1] / (L * H);     // 1760
  const int B_ep   = rows_s / (N * Kshot);      // 32
  const int nqPerEp = rows_q / B_ep;            // 55

  float* ws   = (float*)d_ws;
  float* Ks   = ws; ws += (size_t)rows_s * H;
  float* Kq   = ws; ws += (size_t)rows_q * H;
  float* S2   = ws; ws += (size_t)rows_s * H;
  float* Q2   = ws; ws += (size_t)rows_q * H;
  float* Rbuf = ws; ws += (size_t)M * H;
  float* mem1 = ws; ws += (size_t)M * H;
  float* prot = ws; ws += (size_t)B_ep * N * H;

  // 1) streaming max+LN (HBM-bound, ~440MB total -> ~19us roofline)
  rowmax_ln_kernel<<<rows_s, 256, 0, stream>>>(support, gamma, beta, Ks, L);
  rowmax_ln_kernel<<<rows_q, 256, 0, stream>>>(query, gamma, beta, Kq, L);
  // 2) MANN step for support (WMMA f32): S2 = R+K, keep R[:M] for mem update
  mann_tile_kernel<<<rows_s / 16, 256, 0, stream>>>(Ks, mem0, S2, Rbuf, M);
  // 3) mem1[k] = R[rank(k)] (softmax-of-weights provably dead downstream)
  mem_update_kernel<<<M, 256, 0, stream>>>(mw, Rbuf, mem1);
  // 4) MANN step for query against updated memory
  mann_tile_kernel<<<rows_q / 16, 256, 0, stream>>>(Kq, mem1, Q2, nullptr, M);
  // 5) prototypes + logits/pred
  proto_kernel<<<B_ep * N, 256, 0, stream>>>(S2, prot, Kshot);
  float* out_logits = (float*)d_out;
  float* out_pred = out_logits + (size_t)rows_q * (N + 1);
  logits_kernel<<<rows_q, 256, 0, stream>>>(Q2, prot, out_logits, out_pred,
                                            nqPerEp, N);
}